// TemperGraph_27599459844280
// MI455X (gfx1250) — compile-verified
//
#include <hip/hip_runtime.h>

#define HH    256
#define EDIM  128
#define TT    12
#define OO    3
#define HOPS  4
#define BB    128
#define DD    2048
#define NCLS  10
#define NTOK  1024          // B*D/H
#define NEXP  (TT*OO)       // 36
#define KXA   (HH+EDIM)     // 384

typedef __attribute__((ext_vector_type(2))) float v2f;
typedef __attribute__((ext_vector_type(8))) float v8f;
typedef __attribute__((address_space(3))) float lds_f;

// ---------------- init: state = x, active = 1 ----------------
__global__ void k_init(const float* __restrict__ x, float* __restrict__ state,
                       int* __restrict__ active) {
  int i = blockIdx.x * blockDim.x + threadIdx.x;
  if (i < BB * DD) state[i] = x[i];
  if (i < NTOK) active[i] = 1;
}

// ---------------- zero expert counts ----------------
__global__ void k_zero(int* __restrict__ counts) {
  if (threadIdx.x < NEXP) counts[threadIdx.x] = 0;
}

// ---------------- route: bucket active tokens by expert ----------------
__global__ void k_route(const int* __restrict__ tempers, const int* __restrict__ ops,
                        int* __restrict__ active, int* __restrict__ counts,
                        int* __restrict__ bucket, int hop) {
  int n = blockIdx.x * blockDim.x + threadIdx.x;
  if (n >= NTOK) return;
  int t = tempers[hop * NTOK + n];
  int o = ops[hop * NTOK + n];
  int a = active[n] & (t != TT);
  active[n] = a;
  if (a) {
    int e = t * OO + o;
    int pos = atomicAdd(&counts[e], 1);
    bucket[e * NTOK + pos] = n;
  }
}

// ---------------- expert grouped GEMM via fp32 WMMA ----------------
#define XA_STRIDE 388   // 384 padded: avoids 64-bank alignment conflicts; *4B is 16B-aligned
#define H_STRIDE  260   // 256 padded

__global__ __launch_bounds__(256)
void k_expert(const float* __restrict__ W1, const float* __restrict__ b1,
              const float* __restrict__ W2, const float* __restrict__ b2,
              const float* __restrict__ op_emb,
              const int* __restrict__ counts, const int* __restrict__ bucket,
              float* __restrict__ state)
{
  __shared__ float s_xa[16 * XA_STRIDE];   // [16 tokens][384] concat input
  __shared__ float s_h [16 * H_STRIDE];    // [16 tokens][256] hidden
  __shared__ int   s_tok[16];

  const int e     = blockIdx.x / (NTOK / 16);
  const int mtile = blockIdx.x % (NTOK / 16);
  const int cnt   = counts[e];
  if (mtile * 16 >= cnt) return;           // block-uniform exit: EXEC stays full
  const int rows  = min(16, cnt - mtile * 16);

  const int tid = threadIdx.x;
  if (tid < 16) {
    int idx = mtile * 16 + tid;
    s_tok[tid] = bucket[e * NTOK + ((idx < cnt) ? idx : mtile * 16)];
  }
  __syncthreads();

  // ---- async gather: 16 token state rows -> LDS via direct memory->LDS path
  // (GLOBAL_LOAD_ASYNC_TO_LDS_B128, tracked by ASYNCcnt; bypasses VGPRs)
  for (int idx = tid; idx < 16 * (HH / 4); idx += 256) {   // 1024 float4 chunks
    int i  = idx >> 6;            // token row 0..15
    int j4 = (idx & 63) << 2;     // starting column (multiple of 4)
    const float* g = state + (size_t)s_tok[i] * HH + j4;
    unsigned lds = (unsigned)(unsigned long long)(lds_f*)&s_xa[i * XA_STRIDE + j4];
    asm volatile("global_load_async_to_lds_b128 %0, %1, off"
                 :: "v"(lds), "v"(g) : "memory");
  }
  // emb part [16][128]: identical row for every token of this expert
  for (int idx = tid; idx < 16 * EDIM; idx += 256) {
    int i = idx >> 7, j = idx & 127;
    s_xa[i * XA_STRIDE + HH + j] = op_emb[e * EDIM + j];
  }
  asm volatile("s_wait_asynccnt 0x0" ::: "memory");  // own wave's asyncs done
  __syncthreads();                                   // publish all waves' LDS writes

  const int lane = tid & 31;
  const int wave = tid >> 5;
  const int mrow = lane & 15;                // M (for A) / N (for B/C) sub-index
  const int koff = (lane >> 4) << 1;         // upper half-wave holds K+2,K+3
  const int mhi  = (lane >> 4) << 3;         // C rows: M = r + 8*(lane>=16)

  const float* W1e = W1 + (size_t)e * KXA * HH;
  const float* W2e = W2 + (size_t)e * HH * HH;

  // ---- layer 1: h = relu(xa @ W1[e] + b1[e]) ----
  for (int nt = wave * 2; nt < wave * 2 + 2; ++nt) {
    const int n0 = nt * 16;
    const float bv = b1[e * HH + n0 + mrow];   // bias depends only on N
    v8f acc = {bv, bv, bv, bv, bv, bv, bv, bv};
    #pragma unroll 4
    for (int k0 = 0; k0 < KXA; k0 += 4) {
      v2f a, b;
      a.x = s_xa[mrow * XA_STRIDE + k0 + koff];
      a.y = s_xa[mrow * XA_STRIDE + k0 + koff + 1];
      b.x = W1e[(size_t)(k0 + koff) * HH + n0 + mrow];
      b.y = W1e[(size_t)(k0 + koff + 1) * HH + n0 + mrow];
      acc = __builtin_amdgcn_wmma_f32_16x16x4_f32(
          false, a, false, b, (short)0, acc, false, false);
    }
    #pragma unroll
    for (int r = 0; r < 8; ++r) {
      float v = acc[r];
      s_h[(r + mhi) * H_STRIDE + n0 + mrow] = v > 0.f ? v : 0.f;
    }
  }
  __syncthreads();

  // ---- layer 2: y = relu(h @ W2[e] + b2[e]) -> state (valid rows only) ----
  for (int nt = wave * 2; nt < wave * 2 + 2; ++nt) {
    const int n0 = nt * 16;
    const float bv = b2[e * HH + n0 + mrow];
    v8f acc = {bv, bv, bv, bv, bv, bv, bv, bv};
    #pragma unroll 4
    for (int k0 = 0; k0 < HH; k0 += 4) {
      v2f a, b;
      a.x = s_h[mrow * H_STRIDE + k0 + koff];
      a.y = s_h[mrow * H_STRIDE + k0 + koff + 1];
      b.x = W2e[(size_t)(k0 + koff) * HH + n0 + mrow];
      b.y = W2e[(size_t)(k0 + koff + 1) * HH + n0 + mrow];
      acc = __builtin_amdgcn_wmma_f32_16x16x4_f32(
          false, a, false, b, (short)0, acc, false, false);
    }
    #pragma unroll
    for (int r = 0; r < 8; ++r) {
      int M = r + mhi;
      if (M < rows) {                        // divergence only after all WMMAs
        float v = acc[r];
        state[s_tok[M] * HH + n0 + mrow] = v > 0.f ? v : 0.f;
      }
    }
  }
}

// ---------------- pool + layernorm + task head ----------------
__global__ __launch_bounds__(256)
void k_head(const float* __restrict__ state, const float* __restrict__ ln_g,
            const float* __restrict__ ln_b, const float* __restrict__ Wt,
            const float* __restrict__ bt, float* __restrict__ out)
{
  __shared__ float red[256];
  __shared__ float s_norm[256];
  const int b = blockIdx.x;
  const int h = threadIdx.x;

  float p = 0.f;
  #pragma unroll
  for (int q = 0; q < 8; ++q) p += state[(b * 8 + q) * HH + h];
  p *= 0.125f;

  red[h] = p; __syncthreads();
  for (int s = 128; s > 0; s >>= 1) { if (h < s) red[h] += red[h + s]; __syncthreads(); }
  const float mu = red[0] * (1.f / 256.f);
  __syncthreads();

  const float d = p - mu;
  red[h] = d * d; __syncthreads();
  for (int s = 128; s > 0; s >>= 1) { if (h < s) red[h] += red[h + s]; __syncthreads(); }
  const float var = red[0] * (1.f / 256.f);
  __syncthreads();

  s_norm[h] = d * rsqrtf(var + 1e-5f) * ln_g[h] + ln_b[h];
  __syncthreads();

  for (int c = 0; c < NCLS; ++c) {
    red[h] = s_norm[h] * Wt[h * NCLS + c];
    __syncthreads();
    for (int s = 128; s > 0; s >>= 1) { if (h < s) red[h] += red[h + s]; __syncthreads(); }
    if (h == 0) out[b * NCLS + c] = red[0] + bt[c];
    __syncthreads();
  }
}

extern "C" void kernel_launch(void* const* d_in, const int* in_sizes, int n_in,
                              void* d_out, int out_size, void* d_ws, size_t ws_size,
                              hipStream_t stream) {
  const float* x      = (const float*)d_in[0];
  const float* W1     = (const float*)d_in[1];
  const float* b1     = (const float*)d_in[2];
  const float* W2     = (const float*)d_in[3];
  const float* b2     = (const float*)d_in[4];
  const float* op_emb = (const float*)d_in[5];
  const float* ln_g   = (const float*)d_in[6];
  const float* ln_b   = (const float*)d_in[7];
  const float* Wt     = (const float*)d_in[8];
  const float* bt     = (const float*)d_in[9];
  const int* tempers  = (const int*)d_in[10];
  const int* ops      = (const int*)d_in[11];

  char* ws = (char*)d_ws;
  float* state = (float*)ws;            ws += (size_t)BB * DD * sizeof(float);
  int*   active = (int*)ws;             ws += (size_t)NTOK * sizeof(int);
  int*   counts = (int*)ws;             ws += 64 * sizeof(int);
  int*   bucket = (int*)ws;             // NEXP * NTOK ints (~144 KB)

  k_init<<<(BB * DD + 255) / 256, 256, 0, stream>>>(x, state, active);

  for (int hop = 0; hop < HOPS; ++hop) {
    k_zero<<<1, 64, 0, stream>>>(counts);
    k_route<<<(NTOK + 255) / 256, 256, 0, stream>>>(tempers, ops, active, counts, bucket, hop);
    k_expert<<<NEXP * (NTOK / 16), 256, 0, stream>>>(W1, b1, W2, b2, op_emb,
                                                     counts, bucket, state);
  }

  k_head<<<BB, 256, 0, stream>>>(state, ln_g, ln_b, Wt, bt, (float*)d_out);
}